// SuperpointGCN_7146825581106
// MI455X (gfx1250) — compile-verified
//
#include <hip/hip_runtime.h>
#include <hip/hip_bf16.h>

#define D 128
#define LN_EPS 1e-5f

typedef __attribute__((ext_vector_type(2))) float v2f;
typedef __attribute__((ext_vector_type(8))) float v8f;

// ---------------------------------------------------------------- utilities
__global__ __launch_bounds__(256) void zero_f32_kernel(float* __restrict__ p, long n4) {
    long i = (long)blockIdx.x * blockDim.x + threadIdx.x;
    if (i < n4) ((float4*)p)[i] = make_float4(0.f, 0.f, 0.f, 0.f);
}

__global__ __launch_bounds__(256) void zero_i32_kernel(int* __restrict__ p, int n) {
    int i = blockIdx.x * blockDim.x + threadIdx.x;
    if (i < n) p[i] = 0;
}

__global__ __launch_bounds__(256) void indeg_kernel(const int* __restrict__ dst,
                                                    int* __restrict__ indeg, int E) {
    int i = blockIdx.x * blockDim.x + threadIdx.x;
    if (i < E) atomicAdd(&indeg[dst[i]], 1);
}

// deg[v] = indeg[v] + 1 (outer self loop, w=1) + fill (inner self loop)
__global__ __launch_bounds__(256) void dinv_kernel(const int* __restrict__ indeg,
                                                   float* __restrict__ dinv2,
                                                   float* __restrict__ dinv1, int N) {
    int v = blockIdx.x * blockDim.x + threadIdx.x;
    if (v < N) {
        float d = (float)indeg[v];
        dinv2[v] = rsqrtf(d + 3.0f);  // fill = 2 (improved=True)
        dinv1[v] = rsqrtf(d + 2.0f);  // fill = 1
    }
}

// ---------------------------------------------------------------- GEMM (WMMA f32)
// HS = diag(dinv) * (H @ W).  One wave computes a 16-row x 128-col strip.
// A (16x4 f32): lane l<16 holds A[l][k0], A[l][k0+1]; lane l+16 holds A[l][k0+2..3].
// B (4x16 f32): lane l<16 holds B[k0][n], B[k0+1][n]; lane l+16 holds B[k0+2..3][n].
// C/D (16x16 f32): VGPR j of lane l holds C[j + 8*(l>>4)][l&15].
//
// W is staged in LDS pair-interleaved along K:
//   Wlds[((k>>1)*D + n)*2 + (k&1)] = W[k][n]
// so every B operand is one aligned ds_load_b64 with an immediate offset.
__global__ __launch_bounds__(256) void gemm_hs_kernel(
    const float* __restrict__ H, const float* __restrict__ Wm,
    const float* __restrict__ dinv, float* __restrict__ HS, int N)
{
    __shared__ float Wlds[D * D];
    {
        // work item w = p*D + n, p = K-pair index [0,64), n = column [0,128)
        float2* Wl = (float2*)Wlds;
        for (int w = threadIdx.x; w < (D / 2) * D; w += 256) {
            int p = w >> 7;
            int n = w & (D - 1);
            float2 val;
            val.x = Wm[(2 * p + 0) * D + n];  // coalesced across threads
            val.y = Wm[(2 * p + 1) * D + n];  // coalesced across threads
            Wl[w] = val;                      // contiguous b64 LDS write
        }
    }
    __syncthreads();

    const int lane = threadIdx.x & 31;
    const int wave = threadIdx.x >> 5;
    const int half = lane >> 4;
    const int lr   = lane & 15;

    long r0 = ((long)blockIdx.x * 8 + wave) * 16;
    if (r0 >= N) return;  // wave-uniform: EXEC stays all-ones for WMMA

    long ar = r0 + lr;               // A-row this lane supplies (clamped for tail tile;
    if (ar >= N) ar = N - 1;         // tail stores are masked below)
    const float* __restrict__ Arow = H + ar * D;

    // per-lane LDS base; all inner-loop offsets are compile-time immediates
    const float* __restrict__ Bbase = Wlds + half * (2 * D) + 2 * lr;

    v8f acc[8];
#pragma unroll
    for (int nt = 0; nt < 8; ++nt)
#pragma unroll
        for (int j = 0; j < 8; ++j) acc[nt][j] = 0.0f;

#pragma unroll 4
    for (int k0 = 0; k0 < D; k0 += 4) {
        v2f a = *(const v2f*)(Arow + k0 + 2 * half);  // one global_load_b64
#pragma unroll
        for (int nt = 0; nt < 8; ++nt) {
            v2f b = *(const v2f*)(Bbase + k0 * D + nt * 32);  // one ds_load_b64
            acc[nt] = __builtin_amdgcn_wmma_f32_16x16x4_f32(
                false, a, false, b, (short)0, acc[nt], false, false);
        }
    }

    long  rr[8];
    float dv[8];
    bool  ok[8];
#pragma unroll
    for (int j = 0; j < 8; ++j) {
        rr[j] = r0 + 8 * half + j;
        ok[j] = rr[j] < N;
        dv[j] = ok[j] ? dinv[rr[j]] : 0.0f;
    }
#pragma unroll
    for (int nt = 0; nt < 8; ++nt)
#pragma unroll
        for (int j = 0; j < 8; ++j)
            if (ok[j]) HS[rr[j] * D + nt * 16 + lr] = acc[nt][j] * dv[j];
}

// ---------------------------------------------------------------- edge scatter
// One wave per edge: 32 lanes x float4 = full 128-channel row. Coalesced 512B
// gather from L2-resident HS, 4 f32 atomics per lane into ACC.
__global__ __launch_bounds__(256) void edge_agg_kernel(
    const int* __restrict__ src, const int* __restrict__ dst,
    const float* __restrict__ HS, float* __restrict__ ACC, int E)
{
    long tid = (long)blockIdx.x * blockDim.x + threadIdx.x;
    long e = tid >> 5;
    if (e >= E) return;
    int c = ((int)tid & 31) << 2;
    long s = src[e];
    long d = dst[e];
    float4 v = *(const float4*)(HS + s * D + c);
    float* p = ACC + d * D + c;
    atomicAdd(p + 0, v.x);
    atomicAdd(p + 1, v.y);
    atomicAdd(p + 2, v.z);
    atomicAdd(p + 3, v.w);
}

// ---------------------------------------------------------------- epilogues
__device__ __forceinline__ float wave_sum32(float v) {
#pragma unroll
    for (int off = 16; off >= 1; off >>= 1) v += __shfl_xor(v, off, 32);
    return v;
}

// out = relu(LN(dinv*(acc + selfw*hs) + bias) * gamma + beta); one wave per row
__global__ __launch_bounds__(256) void epilogue_ln_kernel(
    const float* __restrict__ ACC, const float* __restrict__ HS,
    const float* __restrict__ dinv, const float* __restrict__ bias,
    const float* __restrict__ gamma, const float* __restrict__ beta,
    float* __restrict__ OUT, float selfw, int N)
{
    int lane = threadIdx.x & 31;
    long row = (long)blockIdx.x * 8 + (threadIdx.x >> 5);
    if (row >= N) return;
    float dv = dinv[row];
    int c = lane * 4;
    float4 a  = *(const float4*)(ACC + row * D + c);
    float4 h  = *(const float4*)(HS + row * D + c);
    float4 bb = *(const float4*)(bias + c);
    float t0 = dv * (a.x + selfw * h.x) + bb.x;
    float t1 = dv * (a.y + selfw * h.y) + bb.y;
    float t2 = dv * (a.z + selfw * h.z) + bb.z;
    float t3 = dv * (a.w + selfw * h.w) + bb.w;
    float mu = wave_sum32(t0 + t1 + t2 + t3) * (1.0f / 128.0f);
    float d0 = t0 - mu, d1 = t1 - mu, d2 = t2 - mu, d3 = t3 - mu;
    float var = wave_sum32(d0 * d0 + d1 * d1 + d2 * d2 + d3 * d3) * (1.0f / 128.0f);
    float inv = rsqrtf(var + LN_EPS);
    float4 g  = *(const float4*)(gamma + c);
    float4 be = *(const float4*)(beta + c);
    float4 o;
    o.x = fmaxf(d0 * inv * g.x + be.x, 0.0f);
    o.y = fmaxf(d1 * inv * g.y + be.y, 0.0f);
    o.z = fmaxf(d2 * inv * g.z + be.z, 0.0f);
    o.w = fmaxf(d3 * inv * g.w + be.w, 0.0f);
    *(float4*)(OUT + row * D + c) = o;
}

// out = dinv*(acc + selfw*hs) + bias + x   (final layer + residual)
__global__ __launch_bounds__(256) void epilogue_res_kernel(
    const float* __restrict__ ACC, const float* __restrict__ HS,
    const float* __restrict__ dinv, const float* __restrict__ bias,
    const float* __restrict__ X, float* __restrict__ OUT, float selfw, int N)
{
    int lane = threadIdx.x & 31;
    long row = (long)blockIdx.x * 8 + (threadIdx.x >> 5);
    if (row >= N) return;
    float dv = dinv[row];
    int c = lane * 4;
    float4 a  = *(const float4*)(ACC + row * D + c);
    float4 h  = *(const float4*)(HS + row * D + c);
    float4 bb = *(const float4*)(bias + c);
    float4 x  = *(const float4*)(X + row * D + c);
    float4 o;
    o.x = dv * (a.x + selfw * h.x) + bb.x + x.x;
    o.y = dv * (a.y + selfw * h.y) + bb.y + x.y;
    o.z = dv * (a.z + selfw * h.z) + bb.z + x.z;
    o.w = dv * (a.w + selfw * h.w) + bb.w + x.w;
    *(float4*)(OUT + row * D + c) = o;
}

// ---------------------------------------------------------------- launch
extern "C" void kernel_launch(void* const* d_in, const int* in_sizes, int n_in,
                              void* d_out, int out_size, void* d_ws, size_t ws_size,
                              hipStream_t stream)
{
    const float* x   = (const float*)d_in[0];
    const int*   ei  = (const int*)d_in[1];
    const float* W1  = (const float*)d_in[2];
    const float* b1  = (const float*)d_in[3];
    const float* g1  = (const float*)d_in[4];
    const float* be1 = (const float*)d_in[5];
    const float* W2  = (const float*)d_in[6];
    const float* b2  = (const float*)d_in[7];
    const float* g2  = (const float*)d_in[8];
    const float* be2 = (const float*)d_in[9];
    const float* W3  = (const float*)d_in[10];
    const float* b3  = (const float*)d_in[11];

    const int N = in_sizes[0] / D;
    const int E = in_sizes[1] / 2;
    const int* src = ei;
    const int* dst = ei + E;

    char* ws = (char*)d_ws;
    size_t nd = (size_t)N * D;
    float* hs    = (float*)ws; ws += nd * sizeof(float);
    float* acc   = (float*)ws; ws += nd * sizeof(float);
    float* h1    = (float*)ws; ws += nd * sizeof(float);
    int*   indeg = (int*)ws;   ws += (size_t)N * sizeof(int);
    float* dinv2 = (float*)ws; ws += (size_t)N * sizeof(float);
    float* dinv1 = (float*)ws;

    dim3 blk(256);
    int gN = (N + 255) / 256;
    int gE = (E + 255) / 256;
    long nd4 = (long)(nd / 4);
    int gZ = (int)((nd4 + 255) / 256);
    int gRows = (N + 7) / 8;
    int gGemm = (int)((((long)N + 15) / 16 + 7) / 8);
    long eThreads = (long)E * 32;
    int gEdge = (int)((eThreads + 255) / 256);

    // shared degree normalization tables
    zero_i32_kernel<<<gN, blk, 0, stream>>>(indeg, N);
    indeg_kernel<<<gE, blk, 0, stream>>>(dst, indeg, E);
    dinv_kernel<<<gN, blk, 0, stream>>>(indeg, dinv2, dinv1, N);

    // layer 1 (improved: fill=2 -> selfw = 3)
    gemm_hs_kernel<<<gGemm, blk, 0, stream>>>(x, W1, dinv2, hs, N);
    zero_f32_kernel<<<gZ, blk, 0, stream>>>(acc, nd4);
    edge_agg_kernel<<<gEdge, blk, 0, stream>>>(src, dst, hs, acc, E);
    epilogue_ln_kernel<<<gRows, blk, 0, stream>>>(acc, hs, dinv2, b1, g1, be1, h1, 3.0f, N);

    // layer 2 (improved: fill=2 -> selfw = 3)
    gemm_hs_kernel<<<gGemm, blk, 0, stream>>>(h1, W2, dinv2, hs, N);
    zero_f32_kernel<<<gZ, blk, 0, stream>>>(acc, nd4);
    edge_agg_kernel<<<gEdge, blk, 0, stream>>>(src, dst, hs, acc, E);
    epilogue_ln_kernel<<<gRows, blk, 0, stream>>>(acc, hs, dinv2, b2, g2, be2, h1, 3.0f, N);

    // layer 3 (fill=1 -> selfw = 2) + residual, straight to d_out
    gemm_hs_kernel<<<gGemm, blk, 0, stream>>>(h1, W3, dinv1, hs, N);
    zero_f32_kernel<<<gZ, blk, 0, stream>>>(acc, nd4);
    edge_agg_kernel<<<gEdge, blk, 0, stream>>>(src, dst, hs, acc, E);
    epilogue_res_kernel<<<gRows, blk, 0, stream>>>(acc, hs, dinv1, b3, x, (float*)d_out, 2.0f, N);
}